// KERC22BaselineModel_35777077575730
// MI455X (gfx1250) — compile-verified
//
#include <hip/hip_runtime.h>
#include <hip/hip_bf16.h>
#include <math.h>

typedef __attribute__((ext_vector_type(2))) float v2f;
typedef __attribute__((ext_vector_type(8))) float v8f;

// ---------------------------------------------------------------------------
// Degenerate d=1 attention: out[b,s] = sum_t softmax_t(q[b,s]*k[b,t]) * v[b,t]
// One block handles 256 consecutive s for one batch. k,v staged in LDS.
// max_t(q_s*k_t) = q_s>=0 ? q_s*max(k) : q_s*min(k)  (one reduction per block).
// ---------------------------------------------------------------------------
__global__ void attn_d1_kernel(const float* __restrict__ Q,
                               const float* __restrict__ Kp,
                               const float* __restrict__ Vp,
                               float* __restrict__ out,
                               int S, int ostride, int ooffset)
{
    __shared__ float sk[2304];
    __shared__ float sv[2304];
    __shared__ float rmax[256];
    __shared__ float rmin[256];

    const int b   = blockIdx.y;
    const int tid = threadIdx.x;

    const float* krow = Kp + (size_t)b * S;
    const float* vrow = Vp + (size_t)b * S;
    for (int t = tid; t < S; t += 256) { sk[t] = krow[t]; sv[t] = vrow[t]; }
    __syncthreads();

    float lmax = -__builtin_inff(), lmin = __builtin_inff();
    for (int t = tid; t < S; t += 256) {
        float x = sk[t];
        lmax = fmaxf(lmax, x);
        lmin = fminf(lmin, x);
    }
    rmax[tid] = lmax; rmin[tid] = lmin;
    __syncthreads();
    for (int off = 128; off > 0; off >>= 1) {
        if (tid < off) {
            rmax[tid] = fmaxf(rmax[tid], rmax[tid + off]);
            rmin[tid] = fminf(rmin[tid], rmin[tid + off]);
        }
        __syncthreads();
    }
    const float kmax = rmax[0];
    const float kmin = rmin[0];

    const int s = blockIdx.x * 256 + tid;
    if (s < S) {
        const float qs = Q[(size_t)b * S + s];
        const float m  = (qs >= 0.f) ? qs * kmax : qs * kmin;
        float sum = 0.f, acc = 0.f;
        for (int t = 0; t < S; ++t) {
            float e = __expf(qs * sk[t] - m);
            sum += e;
            acc += e * sv[t];
        }
        out[(size_t)b * ostride + ooffset + s] = acc / sum;
    }
}

// ---------------------------------------------------------------------------
// fused[b, 0:768] = sentence[b, :]
// ---------------------------------------------------------------------------
__global__ void copy_sentence_kernel(const float* __restrict__ s,
                                     float* __restrict__ fused)
{
    int i = blockIdx.x * 256 + threadIdx.x;
    if (i >= 32 * 768) return;
    int b = i / 768, c = i % 768;
    fused[(size_t)b * 2304 + c] = s[i];
}

// ---------------------------------------------------------------------------
// fp32 WMMA GEMM: C[M,N] = act(A[M,K] @ B[K,N] + bias[N])
// One wave32 per 16x16 tile, K stepped by 4 via V_WMMA_F32_16X16X4_F32.
// Requires M%16==0, N%16==0, K%4==0. Grid sized so every wave owns a tile
// (EXEC all-ones around the WMMAs).
//   A frag (16x4): lane l -> A[m0+(l&15)][k0+2*(l>>4)+{0,1}]
//   B frag (4x16): lane l -> B[k0+2*(l>>4)+{0,1}][n0+(l&15)]
//   D (16x16):     vgpr r -> row m0+r+8*(l>>4), col n0+(l&15)
// ---------------------------------------------------------------------------
__global__ void gemm_wmma_f32_kernel(const float* __restrict__ A,
                                     const float* __restrict__ B,
                                     const float* __restrict__ bias,
                                     float* __restrict__ C,
                                     int M, int N, int K, int act)
{
    const int wave   = (blockIdx.x * blockDim.x + threadIdx.x) >> 5;
    const int lane   = threadIdx.x & 31;
    const int ntiles = N >> 4;
    const int tot    = (M >> 4) * ntiles;
    if (wave >= tot) return;  // grids are exact; uniform per wave anyway

    const int tm = wave / ntiles;
    const int tn = wave - tm * ntiles;
    const int m0 = tm << 4;
    const int n0 = tn << 4;
    const int half = lane >> 4;
    const int lo   = lane & 15;

    const float* Arow = A + (size_t)(m0 + lo) * K;
    const float* Bcol = B + (n0 + lo);

    v8f acc = {};
    for (int k0 = 0; k0 < K; k0 += 4) {
        const int ka = k0 + 2 * half;
        v2f a;
        a.x = Arow[ka];
        a.y = Arow[ka + 1];
        v2f bfrag;
        bfrag.x = Bcol[(size_t)ka * N];
        bfrag.y = Bcol[(size_t)(ka + 1) * N];
        acc = __builtin_amdgcn_wmma_f32_16x16x4_f32(
            /*neg_a=*/false, a, /*neg_b=*/false, bfrag,
            /*c_mod=*/(short)0, acc, /*reuse_a=*/false, /*reuse_b=*/false);
    }

    const float bv = bias ? bias[n0 + lo] : 0.f;
#pragma unroll
    for (int r = 0; r < 8; ++r) {
        float x = acc[r] + bv;
        if (act == 1) x = fmaxf(x, 0.f);
        C[(size_t)(m0 + r + 8 * half) * N + (n0 + lo)] = x;
    }
}

// ---------------------------------------------------------------------------
// final[b, 0:2304]              = z[b, :]
// final[b, 2304+512*j+n]        = shared_out[3*b+j, n]   (bias already added)
//   shared_out rows (viewed from fused as [96,768]@W_shared):
//   row 3b+0 = sentence_shared, 3b+1 = scene_shared, 3b+2 = speaker_shared
// ---------------------------------------------------------------------------
__global__ void assemble_final_kernel(const float* __restrict__ z,
                                      const float* __restrict__ sh,
                                      float* __restrict__ fin)
{
    int i = blockIdx.x * 256 + threadIdx.x;
    if (i >= 32 * 3840) return;
    int b = i / 3840, c = i % 3840;
    float v;
    if (c < 2304) {
        v = z[(size_t)b * 2304 + c];
    } else {
        int c2 = c - 2304;
        int j  = c2 >> 9;      // /512
        int n  = c2 & 511;
        v = sh[(size_t)(3 * b + j) * 512 + n];
    }
    fin[i] = v;
}

// ---------------------------------------------------------------------------
// out[b,n] = sigmoid(h[b,:] @ W2[:,n] + b2[n]);  32x7 outputs, one block.
// ---------------------------------------------------------------------------
__global__ void head_kernel(const float* __restrict__ h,
                            const float* __restrict__ W2,
                            const float* __restrict__ b2,
                            float* __restrict__ out)
{
    int i = threadIdx.x;
    if (i >= 32 * 7) return;
    int b = i / 7, n = i % 7;
    float acc = b2[n];
    const float* hrow = h + (size_t)b * 256;
    for (int k = 0; k < 256; ++k) acc += hrow[k] * W2[(size_t)k * 7 + n];
    out[i] = 1.f / (1.f + __expf(-acc));
}

// ---------------------------------------------------------------------------
extern "C" void kernel_launch(void* const* d_in, const int* in_sizes, int n_in,
                              void* d_out, int out_size, void* d_ws, size_t ws_size,
                              hipStream_t stream)
{
    const float* sentence = (const float*)d_in[0];   // [32,768]
    const float* target   = (const float*)d_in[1];   // [32,768]
    const float* other    = (const float*)d_in[2];   // [32,768]
    const float* scene_d  = (const float*)d_in[3];   // [32,768]
    const float* scene_s  = (const float*)d_in[4];   // [32,768]
    const float* W_shared = (const float*)d_in[5];   // [768,512]
    const float* b_shared = (const float*)d_in[6];   // [512]
    const float* W1       = (const float*)d_in[7];   // [3840,256]
    const float* b1       = (const float*)d_in[8];   // [256]
    const float* W2       = (const float*)d_in[9];   // [256,7]
    const float* b2       = (const float*)d_in[10];  // [7]
    float* out = (float*)d_out;                      // [32,7]

    // Workspace layout (floats)
    char* ws = (char*)d_ws;
    float* fused      = (float*)(ws);                     // [32,2304] = 294912 B
    float* z          = (float*)(ws + 294912);            // [32,2304] = 294912 B
    float* shared_out = (float*)(ws + 589824);            // [96,512]  = 196608 B
    float* final_buf  = (float*)(ws + 786432);            // [32,3840] = 491520 B
    float* h          = (float*)(ws + 1277952);           // [32,256]  = 32768 B
    (void)ws_size; (void)in_sizes; (void)n_in; (void)out_size;

    // 1. fused[:,0:768] = sentence
    copy_sentence_kernel<<<(32 * 768 + 255) / 256, 256, 0, stream>>>(sentence, fused);

    // 2. speaker_context = attn(other, target, target) -> fused[:,1536:2304]
    attn_d1_kernel<<<dim3(3, 32), 256, 0, stream>>>(other, target, target,
                                                    fused, 768, 2304, 1536);
    // 3. scene_context = attn(scene_sent, scene_sent, scene_desc) -> fused[:,768:1536]
    attn_d1_kernel<<<dim3(3, 32), 256, 0, stream>>>(scene_s, scene_s, scene_d,
                                                    fused, 768, 2304, 768);

    // 4. z = attn(fused, fused, fused), S = 2304
    attn_d1_kernel<<<dim3(9, 32), 256, 0, stream>>>(fused, fused, fused,
                                                    z, 2304, 2304, 0);

    // 5. shared_out[96,512] = (fused viewed [96,768]) @ W_shared + b_shared
    //    tiles = (96/16)*(512/16) = 192; 8 waves/block -> 24 blocks (exact)
    gemm_wmma_f32_kernel<<<24, 256, 0, stream>>>(fused, W_shared, b_shared,
                                                 shared_out, 96, 512, 768, 0);

    // 6. final = [z | sentence_shared | scene_shared | speaker_shared]
    assemble_final_kernel<<<(32 * 3840 + 255) / 256, 256, 0, stream>>>(z, shared_out,
                                                                       final_buf);

    // 7. h = relu(final @ W1 + b1); tiles = (32/16)*(256/16) = 32 -> 4 blocks (exact)
    gemm_wmma_f32_kernel<<<4, 256, 0, stream>>>(final_buf, W1, b1, h,
                                                32, 256, 3840, 1);

    // 8. out = sigmoid(h @ W2 + b2)
    head_kernel<<<1, 256, 0, stream>>>(h, W2, b2, out);
}